// GraphConv_11046655885864
// MI455X (gfx1250) — compile-verified
//
#include <hip/hip_runtime.h>

#define N_NODES 50000
#define E_EDGES 800000
#define F_IN    256
#define F_OUT   128
#define BATCH   2
#define OB      (F_OUT * BATCH)   // 256 flattened columns, c = o*2 + b

#define NT ((N_NODES + 15) / 16)  // 3125 node tiles (exact: 50000 = 16*3125)
#define OT (F_OUT / 16)           // 8 output-feature tiles
#define NT_PER_BLK 8              // waves per block, one node-tile each

typedef float v2f __attribute__((ext_vector_type(2)));
typedef float v8f __attribute__((ext_vector_type(8)));

// ---------------- degree / normalization ----------------

__global__ void deg_init_kernel(float* __restrict__ deg) {
    int i = blockIdx.x * blockDim.x + threadIdx.x;
    if (i < N_NODES) deg[i] = 1.0f;  // self-loop contributes weight 1
}

__global__ void deg_acc_kernel(const int* __restrict__ row,
                               const float* __restrict__ ew,
                               float* __restrict__ deg) {
    for (int e = blockIdx.x * blockDim.x + threadIdx.x; e < E_EDGES;
         e += gridDim.x * blockDim.x) {
        atomicAdd(&deg[row[e]], ew[e]);
    }
}

__global__ void dinv_kernel(const float* __restrict__ deg,
                            float* __restrict__ dinv) {
    int i = blockIdx.x * blockDim.x + threadIdx.x;
    if (i < N_NODES) {
        float d = deg[i];
        dinv[i] = d > 0.0f ? rsqrtf(d) : 0.0f;
    }
}

// ---------------- dense projection via fp32 WMMA ----------------
// xw[n, o*2+b] = sum_f x[b,n,f] * w[f,o]
// One wave computes one 16(M=node) x 16(N=outfeat) tile, K=256 in steps of 4.
// 8 waves/block share (b, o-tile); weight slab staged in LDS (256x16 f32 = 16KB).

__global__ void gemm_wmma_kernel(const float* __restrict__ x,
                                 const float* __restrict__ w,
                                 float* __restrict__ xw) {
    __shared__ float wtile[F_IN * 16];  // wtile[k*16 + j] = w[k, o0+j]

    const int ntb_blocks = (NT + NT_PER_BLK - 1) / NT_PER_BLK;  // 391
    int blk  = blockIdx.x;
    int ntb  = blk % ntb_blocks;
    int rest = blk / ntb_blocks;
    int ot   = rest % OT;
    int b    = rest / OT;
    int o0   = ot * 16;

    // cooperative stage of the 256x16 weight slab
    for (int i = threadIdx.x; i < F_IN * 16; i += blockDim.x) {
        int k = i >> 4;
        int j = i & 15;
        wtile[i] = w[k * F_OUT + o0 + j];
    }
    __syncthreads();

    int wave = threadIdx.x >> 5;
    int lane = threadIdx.x & 31;
    int nt   = ntb * NT_PER_BLK + wave;
    if (nt >= NT) return;  // wave-uniform: EXEC stays all-1s for live waves

    int n0   = nt * 16;
    int half = lane >> 4;   // 0: lanes 0-15, 1: lanes 16-31
    int l    = lane & 15;

    // A-matrix (16x4 f32) lane layout, ISA 7.12.2:
    //   lanes 0-15 : VGPR0=K0, VGPR1=K1 for row M=lane
    //   lanes 16-31: VGPR0=K2, VGPR1=K3 for row M=lane-16
    const float* xrow = x + ((size_t)b * N_NODES + (size_t)(n0 + l)) * F_IN;

    v8f acc = {};
    for (int k = 0; k < F_IN; k += 4) {
        v2f a, bf;
        a.x  = xrow[k + half * 2 + 0];
        a.y  = xrow[k + half * 2 + 1];
        // B-matrix (4x16 f32): row K striped across lanes within a VGPR;
        // lanes 0-15 hold K={0,1}, lanes 16-31 hold K={2,3}, col N=l
        bf.x = wtile[(k + half * 2 + 0) * 16 + l];
        bf.y = wtile[(k + half * 2 + 1) * 16 + l];
        acc = __builtin_amdgcn_wmma_f32_16x16x4_f32(
            /*neg_a=*/false, a, /*neg_b=*/false, bf,
            /*c_mod=*/(short)0, acc, /*reuse_a=*/false, /*reuse_b=*/false);
    }

    // C/D layout: VGPR v -> rows M=v (lanes 0-15) and M=v+8 (lanes 16-31), col N=l
    #pragma unroll
    for (int v = 0; v < 8; ++v) {
        int m = v + half * 8;
        xw[(size_t)(n0 + m) * OB + (size_t)(o0 + l) * 2 + b] = acc[v];
    }
}

// ---------------- output init: bias + self-loop term ----------------
// out[b,n,o] = bias[o] + dinv[n]^2 * xw[n, o*2+b]

__global__ void out_init_kernel(const float* __restrict__ dinv,
                                const float* __restrict__ xw,
                                const float* __restrict__ bias,
                                float* __restrict__ out) {
    size_t i = (size_t)blockIdx.x * blockDim.x + threadIdx.x;
    const size_t total = (size_t)BATCH * N_NODES * F_OUT;
    if (i >= total) return;
    int o    = (int)(i % F_OUT);
    size_t t = i / F_OUT;
    int n    = (int)(t % N_NODES);
    int b    = (int)(t / N_NODES);
    float d  = dinv[n];
    out[i] = bias[o] + d * d * xw[(size_t)n * OB + (size_t)o * 2 + b];
}

// ---------------- edge scatter (SpMM) ----------------
// out[b, row, o] += dinv[row]*ew*dinv[col] * xw[col, o*2+b]
// One edge per block iteration; 256 threads cover the 256 flattened columns.

__global__ void edge_scatter_kernel(const int* __restrict__ row,
                                    const int* __restrict__ col,
                                    const float* __restrict__ ew,
                                    const float* __restrict__ dinv,
                                    const float* __restrict__ xw,
                                    float* __restrict__ out) {
    int c = threadIdx.x;      // 0..255
    int o = c >> 1;
    int b = c & 1;
    for (int e = blockIdx.x; e < E_EDGES; e += gridDim.x) {
        int r  = row[e];      // block-uniform loads
        int cl = col[e];
        float lap = dinv[r] * ew[e] * dinv[cl];
        float val = lap * xw[(size_t)cl * OB + c];  // coalesced 1KB row gather
        atomicAdd(&out[((size_t)b * N_NODES + r) * F_OUT + o], val);
    }
}

// ---------------- launch ----------------

extern "C" void kernel_launch(void* const* d_in, const int* in_sizes, int n_in,
                              void* d_out, int out_size, void* d_ws, size_t ws_size,
                              hipStream_t stream) {
    const float* x    = (const float*)d_in[0];  // [2, 50000, 256]
    const float* w    = (const float*)d_in[1];  // [256, 128]
    const float* bias = (const float*)d_in[2];  // [128]
    const float* ew   = (const float*)d_in[3];  // [800000]
    const int*   eidx = (const int*)d_in[4];    // [2, 800000]
    const int*   row  = eidx;
    const int*   col  = eidx + E_EDGES;
    float* out = (float*)d_out;

    // workspace: xw [N,256] fp32 (51.2MB) | deg [N] | dinv [N]
    float* xw   = (float*)d_ws;
    float* deg  = xw + (size_t)N_NODES * OB;
    float* dinv = deg + N_NODES;

    deg_init_kernel<<<(N_NODES + 255) / 256, 256, 0, stream>>>(deg);
    deg_acc_kernel<<<1024, 256, 0, stream>>>(row, ew, deg);
    dinv_kernel<<<(N_NODES + 255) / 256, 256, 0, stream>>>(deg, dinv);

    const int ntb_blocks = (NT + NT_PER_BLK - 1) / NT_PER_BLK;  // 391
    gemm_wmma_kernel<<<ntb_blocks * OT * BATCH, 256, 0, stream>>>(x, w, xw);

    size_t total = (size_t)BATCH * N_NODES * F_OUT;
    out_init_kernel<<<(unsigned)((total + 255) / 256), 256, 0, stream>>>(dinv, xw, bias, out);

    edge_scatter_kernel<<<8192, 256, 0, stream>>>(row, col, ew, dinv, xw, out);
}